// GraphEncoder_23605140259146
// MI455X (gfx1250) — compile-verified
//
#include <hip/hip_runtime.h>

#define NODES  50000
#define EDGES  600000
#define FEAT   128
#define NGRAPH 256
#define MT_PER_BLK 5                 // 5 M-tiles of 16 rows = 80 rows/block; 50000 = 625*80
#define ROWS_PER_BLK (MT_PER_BLK * 16)
#define A_LDS_STRIDE 132             // 128 + 4 pad: conflict-free frag reads, 528B = 33*16B aligned

typedef float v2f __attribute__((ext_vector_type(2)));
typedef float v8f __attribute__((ext_vector_type(8)));

// ---------------- degree: deg = 1 + sum over edges at dst; then rsqrt ----------------
__global__ void deg_init_kernel(float* __restrict__ deg) {
    int i = blockIdx.x * blockDim.x + threadIdx.x;
    if (i < NODES) deg[i] = 1.0f;
}

__global__ void deg_accum_kernel(const int* __restrict__ ei, float* __restrict__ deg) {
    int e = blockIdx.x * blockDim.x + threadIdx.x;
    if (e < EDGES) atomicAdd(&deg[ei[EDGES + e]], 1.0f);
}

__global__ void deg_finish_kernel(float* __restrict__ deg) {
    int i = blockIdx.x * blockDim.x + threadIdx.x;
    if (i < NODES) deg[i] = rsqrtf(deg[i]);   // in place: deg -> deg^{-1/2}
}

// ---------------- GEMM: H = (relu?)(X) @ W  via V_WMMA_F32_16X16X4_F32 ----------------
// grid.x = NODES/80; 256 threads = 8 waves; wave w owns cols [16w, 16w+16).
// Each wave computes 5 stacked 16x16 output tiles, reusing each B fragment 5x.
__global__ __launch_bounds__(256)
void gcn_gemm_wmma_kernel(const float* __restrict__ X, const float* __restrict__ W,
                          float* __restrict__ Hout, int relu_in) {
    __shared__ float As[ROWS_PER_BLK * A_LDS_STRIDE];

    const int m0  = blockIdx.x * ROWS_PER_BLK;
    const int tid = threadIdx.x;

    // Stage 80x128 A tile into LDS as float4 (ReLU of previous layer fused in).
    for (int i = tid; i < ROWS_PER_BLK * (FEAT / 4); i += 256) {
        const int r  = i >> 5;           // i / 32
        const int c4 = i & 31;           // float4 column
        float4 v = ((const float4*)X)[(size_t)(m0 + r) * (FEAT / 4) + c4];
        if (relu_in) {
            v.x = fmaxf(v.x, 0.0f); v.y = fmaxf(v.y, 0.0f);
            v.z = fmaxf(v.z, 0.0f); v.w = fmaxf(v.w, 0.0f);
        }
        *(float4*)&As[r * A_LDS_STRIDE + c4 * 4] = v;
    }
    __syncthreads();

    const int lane = tid & 31;
    const int wave = tid >> 5;
    const int n0   = wave * 16;
    const int half = lane >> 4;          // 0: lanes 0-15, 1: lanes 16-31
    const int lm   = lane & 15;

    v8f acc[MT_PER_BLK];
#pragma unroll
    for (int mt = 0; mt < MT_PER_BLK; ++mt) acc[mt] = (v8f){};

#pragma unroll 4
    for (int k0 = 0; k0 < FEAT; k0 += 4) {
        const int kb = k0 + 2 * half;
        // B 4x16 frag (row-striped): lane holds W[kb+0, n0+lm], W[kb+1, n0+lm]
        v2f b;
        b.x = W[(kb + 0) * FEAT + n0 + lm];
        b.y = W[(kb + 1) * FEAT + n0 + lm];
#pragma unroll
        for (int mt = 0; mt < MT_PER_BLK; ++mt) {
            // A 16x4 frag (ISA layout): lane holds A[row, kb+0], A[row, kb+1]
            v2f a;
            a.x = As[(mt * 16 + lm) * A_LDS_STRIDE + kb + 0];
            a.y = As[(mt * 16 + lm) * A_LDS_STRIDE + kb + 1];
            acc[mt] = __builtin_amdgcn_wmma_f32_16x16x4_f32(
                /*neg_a=*/false, a, /*neg_b=*/false, b,
                /*c_mod=*/(short)0, acc[mt], /*reuse_a=*/false, /*reuse_b=*/false);
        }
    }

    // D layout: VGPR v holds row m = v + 8*half, col n0+lm
#pragma unroll
    for (int mt = 0; mt < MT_PER_BLK; ++mt) {
#pragma unroll
        for (int v = 0; v < 8; ++v) {
            Hout[(size_t)(m0 + mt * 16 + v + 8 * half) * FEAT + n0 + lm] = acc[mt][v];
        }
    }
}

// ---------------- agg init: self-loop term + bias ----------------
__global__ void self_bias_kernel(const float* __restrict__ h, const float* __restrict__ dis,
                                 const float* __restrict__ bias, float* __restrict__ agg) {
    int i = blockIdx.x * blockDim.x + threadIdx.x;   // over NODES*FEAT
    int node = i >> 7, f = i & 127;
    float d = dis[node];
    agg[i] = h[i] * d * d + bias[f];
}

// ---------------- edge scatter: agg[dst] += h[src] * dis[src]*dis[dst] ----------------
// 128 threads per edge, 2 edges per 256-thread block
__global__ __launch_bounds__(256)
void edge_scatter_kernel(const int* __restrict__ ei, const float* __restrict__ h,
                         const float* __restrict__ dis, float* __restrict__ agg) {
    int t = blockIdx.x * blockDim.x + threadIdx.x;
    int e = t >> 7, f = t & 127;
    if (e < EDGES) {
        int s = ei[e];
        int d = ei[EDGES + e];
        float w = dis[s] * dis[d];
        atomicAdd(&agg[(size_t)d * FEAT + f], h[(size_t)s * FEAT + f] * w);
    }
}

// ---------------- pooling ----------------
__global__ void pool_zero_kernel(float* __restrict__ sums, float* __restrict__ cnts) {
    int i = blockIdx.x * blockDim.x + threadIdx.x;
    if (i < NGRAPH * FEAT) sums[i] = 0.0f;
    if (i < NGRAPH) cnts[i] = 0.0f;
}

__global__ void pool_accum_kernel(const float* __restrict__ agg, const int* __restrict__ batch,
                                  float* __restrict__ sums, float* __restrict__ cnts) {
    int i = blockIdx.x * blockDim.x + threadIdx.x;   // over NODES*FEAT
    int node = i >> 7, f = i & 127;
    int g = batch[node];
    float v = fmaxf(agg[i], 0.0f);                   // fused final-layer ReLU
    atomicAdd(&sums[g * FEAT + f], v);
    if (f == 0) atomicAdd(&cnts[g], 1.0f);
}

__global__ void pool_finish_kernel(const float* __restrict__ sums, const float* __restrict__ cnts,
                                   float* __restrict__ out) {
    int i = blockIdx.x * blockDim.x + threadIdx.x;   // over NGRAPH*FEAT
    int g = i >> 7;
    out[i] = sums[i] / fmaxf(cnts[g], 1.0f);
}

extern "C" void kernel_launch(void* const* d_in, const int* in_sizes, int n_in,
                              void* d_out, int out_size, void* d_ws, size_t ws_size,
                              hipStream_t stream) {
    const float* x  = (const float*)d_in[0];
    const float* W1 = (const float*)d_in[1];
    const float* b1 = (const float*)d_in[2];
    const float* W2 = (const float*)d_in[3];
    const float* b2 = (const float*)d_in[4];
    const float* W3 = (const float*)d_in[5];
    const float* b3 = (const float*)d_in[6];
    const int*   ei = (const int*)d_in[7];      // edge_index [2, E]
    const int*   bi = (const int*)d_in[8];      // batch_idx  [N]
    float* out = (float*)d_out;

    // workspace carve-up (all offsets 256B-aligned)
    char* wsb = (char*)d_ws;
    size_t off = 0;
    float* dis  = (float*)(wsb + off); off += ((size_t)NODES * 4 + 255) & ~(size_t)255;
    float* hbuf = (float*)(wsb + off); off += (size_t)NODES * FEAT * 4;      // 25.6 MB
    float* abuf = (float*)(wsb + off); off += (size_t)NODES * FEAT * 4;
    float* sums = (float*)(wsb + off); off += (size_t)NGRAPH * FEAT * 4;
    float* cnts = (float*)(wsb + off); off += (size_t)NGRAPH * 4;
    (void)ws_size; (void)in_sizes; (void)n_in; (void)out_size;

    // degrees -> deg^{-1/2}
    deg_init_kernel<<<(NODES + 255) / 256, 256, 0, stream>>>(dis);
    deg_accum_kernel<<<(EDGES + 255) / 256, 256, 0, stream>>>(ei, dis);
    deg_finish_kernel<<<(NODES + 255) / 256, 256, 0, stream>>>(dis);

    const float* Ws[3] = {W1, W2, W3};
    const float* bs[3] = {b1, b2, b3};
    const int gemm_blocks = NODES / ROWS_PER_BLK;    // 625, exact
    const int nf_blocks   = NODES * FEAT / 256;      // 25000
    const int edge_blocks = (EDGES + 1) / 2;         // 2 edges per block

    const float* xin = x;
    for (int layer = 0; layer < 3; ++layer) {
        gcn_gemm_wmma_kernel<<<gemm_blocks, 256, 0, stream>>>(xin, Ws[layer], hbuf,
                                                              /*relu_in=*/layer > 0 ? 1 : 0);
        self_bias_kernel<<<nf_blocks, 256, 0, stream>>>(hbuf, dis, bs[layer], abuf);
        edge_scatter_kernel<<<edge_blocks, 256, 0, stream>>>(ei, hbuf, dis, abuf);
        xin = abuf;   // next layer reads agg (ReLU fused into GEMM A-load / pooling)
    }

    pool_zero_kernel<<<(NGRAPH * FEAT + 255) / 256, 256, 0, stream>>>(sums, cnts);
    pool_accum_kernel<<<nf_blocks, 256, 0, stream>>>(abuf, bi, sums, cnts);
    pool_finish_kernel<<<(NGRAPH * FEAT) / 256, 256, 0, stream>>>(sums, cnts, out);
}